// MallLoss_1322849927791
// MI455X (gfx1250) — compile-verified
//
#include <hip/hip_runtime.h>

#define BS_ROWS 262144
#define FEAT 128
#define ALPHA 0.3f
#define NB 256                               // partial blocks per input
#define ROWS_PER_BLOCK (BS_ROWS / NB)        // 1024
#define NWAVES 8
#define ROWS_PER_WAVE (ROWS_PER_BLOCK / NWAVES) // 128

typedef float vf4 __attribute__((ext_vector_type(4)));
typedef float v2f __attribute__((ext_vector_type(2)));
typedef float v8f __attribute__((ext_vector_type(8)));

// Pass 1: per-input streaming reduction (HBM-bandwidth bound: 640 MB total).
// Each wave owns 128 consecutive rows; lane L owns columns [4L, 4L+3].
// One global_load_b128 (non-temporal) covers one full 128-col row per wave.
__global__ __launch_bounds__(256) void mall_partial_kernel(
    const float* __restrict__ x,   const float* __restrict__ nx,
    const float* __restrict__ ni,  const float* __restrict__ nx1,
    const float* __restrict__ ni1,
    float* __restrict__ ws_col,    // [5][NB][FEAT]
    float* __restrict__ ws_sq)     // [5][NB]
{
    const int v = blockIdx.y;
    const float* in;
    switch (v) {
        case 0: in = x;   break;
        case 1: in = nx;  break;
        case 2: in = ni;  break;
        case 3: in = nx1; break;
        default: in = ni1; break;
    }
    const int tid  = threadIdx.x;
    const int lane = tid & 31;
    const int wave = tid >> 5;

    const size_t rowStart = (size_t)blockIdx.x * ROWS_PER_BLOCK
                          + (size_t)wave * ROWS_PER_WAVE;
    const vf4* p = (const vf4*)in + rowStart * (FEAT / 4) + lane;

    vf4 acc = {0.0f, 0.0f, 0.0f, 0.0f};
    float sq = 0.0f;
#pragma unroll 4
    for (int i = 0; i < ROWS_PER_WAVE; ++i) {
        vf4 d = __builtin_nontemporal_load(p + (size_t)i * (FEAT / 4));
        acc += d;
        sq = fmaf(d.x, d.x, sq);
        sq = fmaf(d.y, d.y, sq);
        sq = fmaf(d.z, d.z, sq);
        sq = fmaf(d.w, d.w, sq);
    }

    __shared__ float s_col[NWAVES][FEAT];
    __shared__ float s_sq[NWAVES];
    s_col[wave][lane * 4 + 0] = acc.x;
    s_col[wave][lane * 4 + 1] = acc.y;
    s_col[wave][lane * 4 + 2] = acc.z;
    s_col[wave][lane * 4 + 3] = acc.w;

    // deterministic in-wave tree reduction of the sum-of-squares (wave32)
    for (int off = 16; off > 0; off >>= 1) sq += __shfl_down(sq, off, 32);
    if (lane == 0) s_sq[wave] = sq;
    __syncthreads();

    if (tid < FEAT) {
        float t = 0.0f;
        for (int w = 0; w < NWAVES; ++w) t += s_col[w][tid];   // fixed order
        ws_col[((size_t)v * NB + blockIdx.x) * FEAT + tid] = t;
    }
    if (tid == 0) {
        float t = 0.0f;
        for (int w = 0; w < NWAVES; ++w) t += s_sq[w];         // fixed order
        ws_sq[v * NB + blockIdx.x] = t;
    }
}

// Pass 2: single block. Reduce partials, then WMMA Gram matrix m_v . c_j,
// apply the coefficient table, emit loss + centers.
// LDS operand tiles are padded to 16 rows of zeros so the WMMA operand
// gather is unconditional (no exec-mask churn, ds_load_b64 per K-pair).
__global__ __launch_bounds__(256) void mall_final_kernel(
    const float* __restrict__ ws_col, const float* __restrict__ ws_sq,
    const float* __restrict__ centers, float* __restrict__ out)
{
    __shared__ float s_m[16][FEAT];  // column sums per input (rows 5..15 = 0)
    __shared__ float s_c[16][FEAT];  // centers               (rows 5..15 = 0)
    __shared__ float s_S[5];         // sum of squares per input
    __shared__ float s_C2[5];        // ||c_j||^2
    __shared__ float s_dot[25];      // m_v . c_j
    const int tid = threadIdx.x;

    // second output of the tuple: centers pass-through
    for (int i = tid; i < 5 * FEAT; i += 256) out[1 + i] = centers[i];

    // fill operand tiles: rows 0..4 = data, rows 5..15 = zero padding
    for (int item = tid; item < 16 * FEAT; item += 256) {
        const int v = item >> 7;
        const int c = item & (FEAT - 1);
        float t = 0.0f;
        float cv = 0.0f;
        if (v < 5) {
            const float* pp = ws_col + (size_t)v * NB * FEAT + c;
            for (int b = 0; b < NB; ++b) t += pp[(size_t)b * FEAT]; // fixed order
            cv = centers[v * FEAT + c];
        }
        s_m[v][c] = t;
        s_c[v][c] = cv;
    }
    if (tid < 5) {
        float t = 0.0f;
        for (int b = 0; b < NB; ++b) t += ws_sq[tid * NB + b];
        s_S[tid] = t;
    }
    __syncthreads();

    // ||c_j||^2 on wave 1 (co-executes with wave-0 WMMA)
    if (tid >= 32 && tid < 37) {
        const int j = tid - 32;
        float t = 0.0f;
        for (int c = 0; c < FEAT; ++c) t = fmaf(s_c[j][c], s_c[j][c], t);
        s_C2[j] = t;
    }

    // Wave 0 (EXEC all ones): D = A(16x4) x B(4x16) + C accumulated over K=128.
    // A 16x4 f32 layout: lanes 0-15 -> K=k0,k0+1 ; lanes 16-31 -> K=k0+2,k0+3
    // (ISA 7.12.2); B 4x16 mirrors with N = lane%16. D[m][n] = m_m . c_n.
    if (tid < 32) {
        const int lane = tid;
        const int mn = lane & 15;            // M row for A, N col for B
        const int kh = (lane >> 4) * 2;      // K half-select per lane group
        v8f dacc = {};
        for (int k0 = 0; k0 < FEAT; k0 += 4) {
            const int kk = k0 + kh;          // even -> 8B-aligned LDS b64 load
            v2f a = *(const v2f*)&s_m[mn][kk];
            v2f b = *(const v2f*)&s_c[mn][kk];
            dacc = __builtin_amdgcn_wmma_f32_16x16x4_f32(
                false, a, false, b, (short)0, dacc, false, false);
        }
        // D layout: VGPR r, lanes 0-15 -> D[M=r][N=lane]
        if (lane < 5) {
            s_dot[0 * 5 + lane] = dacc[0];
            s_dot[1 * 5 + lane] = dacc[1];
            s_dot[2 * 5 + lane] = dacc[2];
            s_dot[3 * 5 + lane] = dacc[3];
            s_dot[4 * 5 + lane] = dacc[4];
        }
    }
    __syncthreads();

    if (tid == 0) {
        const float a = ALPHA;
        const float w[5][5] = {
            { 1.0f, -a,   -a,   -a,   -a   },   // x
            { -a,   3.0f, -a,   -a,   0.0f },   // nx
            { -a,   -a,   0.0f, 3.0f, -a   },   // ni
            { -a,   -a,   3.0f, 0.0f, -a   },   // nx1
            { -a,   0.0f, -a,   -a,   3.0f },   // ni1
        };
        float dist = 0.0f;
        for (int v = 0; v < 5; ++v) {
            float Wv = 0.0f;
            for (int j = 0; j < 5; ++j) Wv += w[v][j];
            dist += Wv * s_S[v];
        }
        for (int j = 0; j < 5; ++j) {
            float cw = 0.0f;
            for (int v = 0; v < 5; ++v) cw += w[v][j];
            dist += (float)BS_ROWS * cw * s_C2[j];
        }
        for (int v = 0; v < 5; ++v)
            for (int j = 0; j < 5; ++j)
                dist -= 2.0f * w[v][j] * s_dot[v * 5 + j];
        out[0] = fmaxf(dist, 0.0f);
    }
}

extern "C" void kernel_launch(void* const* d_in, const int* in_sizes, int n_in,
                              void* d_out, int out_size, void* d_ws, size_t ws_size,
                              hipStream_t stream) {
    const float* x       = (const float*)d_in[0];
    const float* nx      = (const float*)d_in[1];
    const float* ni      = (const float*)d_in[2];
    const float* nx1     = (const float*)d_in[3];
    const float* ni1     = (const float*)d_in[4];
    const float* centers = (const float*)d_in[5];
    float* out = (float*)d_out;

    float* ws_col = (float*)d_ws;                       // 5*NB*FEAT floats
    float* ws_sq  = ws_col + (size_t)5 * NB * FEAT;     // 5*NB floats

    dim3 grid(NB, 5);
    mall_partial_kernel<<<grid, 256, 0, stream>>>(x, nx, ni, nx1, ni1, ws_col, ws_sq);
    mall_final_kernel<<<dim3(1), 256, 0, stream>>>(ws_col, ws_sq, centers, out);
}